// Model_8907762172297
// MI455X (gfx1250) — compile-verified
//
#include <hip/hip_runtime.h>

// ---------------------------------------------------------------------------
// Problem constants (from the reference)
// ---------------------------------------------------------------------------
#define DMODEL 128
#define NMOD   8
#define PLEN   16
#define BATCH  128
#define CHAN   21
#define TLEN   2048
#define LPAT   (TLEN / PLEN)        // 128
#define LPM    (LPAT / NMOD)        // 16 patches per model
#define BC     (BATCH * CHAN)       // 2688
#define NTOT   (BC * TLEN)          // 5,505,024 elements per (B,C,T) tensor

typedef float v2f __attribute__((ext_vector_type(2)));
typedef float v8f __attribute__((ext_vector_type(8)));

// ---------------------------------------------------------------------------
// fp32 WMMA 16x16x4 helper.
// Operand layout (ISA 7.12.2, wave32):
//   A (16x4):  lane L: M = L&15, holds K = k0, k0+1 in .x/.y, k0 = 2*(L>>4)
//   B (4x16):  lane L: N = L&15, holds K = k0, k0+1 in .x/.y, k0 = 2*(L>>4)
//   C/D:       VGPR v: M = v + 8*(L>>4), N = L&15
// ---------------------------------------------------------------------------
__device__ __forceinline__ v8f wmma_f32(v2f a, v2f b, v8f c) {
  return __builtin_amdgcn_wmma_f32_16x16x4_f32(
      /*neg_a=*/false, a, /*neg_b=*/false, b,
      /*c_mod=*/(short)0, c, /*reuse_a=*/false, /*reuse_b=*/false);
}

__device__ __forceinline__ float gelu_tanh(float v) {
  // jax.nn.gelu default (approximate=True)
  const float k0 = 0.7978845608028654f;   // sqrt(2/pi)
  const float k1 = 0.044715f;
  float u = k0 * (v + k1 * v * v * v);
  return 0.5f * v * (1.0f + tanhf(u));
}

// ---------------------------------------------------------------------------
// Kernel 1: fused embed -> enc MLP (gelu) -> enc -> decoder -> scatter.
// grid.x = BC tiles (one (b,c) pair, 16 patch-rows), grid.y = model n.
// 256 threads = 8 wave32; wave w owns output columns [w*16, w*16+16).
// ---------------------------------------------------------------------------
__global__ __launch_bounds__(256) void fused_encdec_kernel(
    const float* __restrict__ xin,      // (B,C,T)
    const float* __restrict__ emb_W,    // (P, D)
    const float* __restrict__ emb_b,    // (D)
    const float* __restrict__ enc_W1,   // (N, D, D)
    const float* __restrict__ enc_b1,   // (N, D)
    const float* __restrict__ enc_W2,   // (N, D, D)
    const float* __restrict__ enc_b2,   // (N, D)
    const float* __restrict__ dec_W,    // (N, D, P)
    const float* __restrict__ dec_b,    // (N, P)
    float* __restrict__ rec)            // (B,C,T)
{
  __shared__ float sA[16 * DMODEL];          // 8 KB: patches / e / h1 / h2
  __shared__ float sPart[NMOD * 16 * PLEN];  // 8 KB: decoder split-K partials

  const int bc   = blockIdx.x;               // (b*C + c)
  const int n    = blockIdx.y;               // model
  const int tid  = threadIdx.x;
  const int w    = tid >> 5;                 // wave id 0..7
  const int lane = tid & 31;
  const int lm   = lane & 15;                // M (A) / N (B) index
  const int k0   = (lane >> 4) << 1;         // K phase: 0 or 2
  const int rh   = (lane >> 4) << 3;         // C-row half offset: 0 or 8

  const float* W1 = enc_W1 + (size_t)n * DMODEL * DMODEL;
  const float* W2 = enc_W2 + (size_t)n * DMODEL * DMODEL;
  const float* dW = dec_W  + (size_t)n * DMODEL * PLEN;
  const size_t base = (size_t)bc * TLEN;

  // ---- stage patches: sA[lo][p] (row stride DMODEL), t = lo*128 + n*16 + p
  {
    int m = tid >> 4, p = tid & 15;          // 256 threads -> 16x16
    sA[m * DMODEL + p] = xin[base + (size_t)(m * 128 + n * 16 + p)];
  }
  __syncthreads();

  const int col = w * 16 + lm;               // this lane's output column in D

  // ---- embed: e(16x128) = patches(16x16) @ emb_W(16x128) + emb_b
  v8f acc = {};
  #pragma unroll
  for (int ki = 0; ki < 4; ++ki) {
    int k = ki * 4 + k0;
    v2f a, b;
    a.x = sA[lm * DMODEL + k];     a.y = sA[lm * DMODEL + k + 1];
    b.x = emb_W[k * DMODEL + col]; b.y = emb_W[(k + 1) * DMODEL + col];
    acc = wmma_f32(a, b, acc);
  }
  {
    float bias = emb_b[col];
    #pragma unroll
    for (int v = 0; v < 8; ++v) acc[v] += bias;
  }
  __syncthreads();
  #pragma unroll
  for (int v = 0; v < 8; ++v) sA[(v + rh) * DMODEL + col] = acc[v];
  __syncthreads();

  // ---- encoder layer 1: h1 = gelu(e @ W1 + b1)
  v8f h = {};
  #pragma unroll 4
  for (int kk = 0; kk < DMODEL; kk += 4) {
    int k = kk + k0;
    v2f a, b;
    a.x = sA[lm * DMODEL + k];      a.y = sA[lm * DMODEL + k + 1];
    b.x = W1[k * DMODEL + col];     b.y = W1[(k + 1) * DMODEL + col];
    h = wmma_f32(a, b, h);
  }
  {
    float bias = enc_b1[n * DMODEL + col];
    #pragma unroll
    for (int v = 0; v < 8; ++v) h[v] = gelu_tanh(h[v] + bias);
  }
  __syncthreads();
  #pragma unroll
  for (int v = 0; v < 8; ++v) sA[(v + rh) * DMODEL + col] = h[v];
  __syncthreads();

  // ---- encoder layer 2: h2 = h1 @ W2 + b2
  v8f h2 = {};
  #pragma unroll 4
  for (int kk = 0; kk < DMODEL; kk += 4) {
    int k = kk + k0;
    v2f a, b;
    a.x = sA[lm * DMODEL + k];      a.y = sA[lm * DMODEL + k + 1];
    b.x = W2[k * DMODEL + col];     b.y = W2[(k + 1) * DMODEL + col];
    h2 = wmma_f32(a, b, h2);
  }
  {
    float bias = enc_b2[n * DMODEL + col];
    #pragma unroll
    for (int v = 0; v < 8; ++v) h2[v] += bias;
  }
  __syncthreads();
  #pragma unroll
  for (int v = 0; v < 8; ++v) sA[(v + rh) * DMODEL + col] = h2[v];
  __syncthreads();

  // ---- decoder, split-K over waves: wave w does K slice [w*16, w*16+16)
  v8f d = {};
  #pragma unroll
  for (int ki = 0; ki < 4; ++ki) {
    int k = w * 16 + ki * 4 + k0;
    v2f a, b;
    a.x = sA[lm * DMODEL + k];  a.y = sA[lm * DMODEL + k + 1];
    b.x = dW[k * PLEN + lm];    b.y = dW[(k + 1) * PLEN + lm];
    d = wmma_f32(a, b, d);
  }
  #pragma unroll
  for (int v = 0; v < 8; ++v)
    sPart[(w * 16 + (v + rh)) * PLEN + lm] = d[v];
  __syncthreads();

  // ---- reduce partials + bias, scatter with t = (lo*16 + p)*8 + n
  {
    int m = tid >> 4, p = tid & 15;
    float s = dec_b[n * PLEN + p];
    #pragma unroll
    for (int ww = 0; ww < NMOD; ++ww)
      s += sPart[(ww * 16 + m) * PLEN + p];
    rec[base + (size_t)((m * 16 + p) * 8 + n)] = s;
  }
}

// ---------------------------------------------------------------------------
// Kernel 2: mapper GEMM  y_hat(2688x2048) = x_rec @ map_W + map_b,
// fused with sum |y_hat - y_rec| (pred-loss numerator).
// grid = (168 row tiles, 16 col tiles); 256 threads = 8 waves;
// tile = 16 rows x 128 cols, K = 2048 staged through LDS in chunks of 64.
// ---------------------------------------------------------------------------
__global__ __launch_bounds__(256) void map_loss_kernel(
    const float* __restrict__ x_rec,    // (BC, T)
    const float* __restrict__ map_W,    // (T, T)
    const float* __restrict__ map_b,    // (T)
    const float* __restrict__ y_rec,    // (BC, T)
    float* __restrict__ y_hat,          // (BC, T)
    double* __restrict__ pred_acc)
{
  __shared__ float sA[16 * 64];          // 4 KB A stage
  __shared__ float sred[256];

  const int tid  = threadIdx.x;
  const int w    = tid >> 5;
  const int lane = tid & 31;
  const int lm   = lane & 15;
  const int k0   = (lane >> 4) << 1;
  const int rh   = (lane >> 4) << 3;

  const int rowBase = blockIdx.x * 16;
  const int col     = blockIdx.y * 128 + w * 16 + lm;

  v8f acc = {};
  for (int kk = 0; kk < TLEN; kk += 64) {
    // cooperative A stage: 16 rows x 64 cols, float4 per thread
    {
      int lin = tid * 4;
      int r = lin >> 6, j = lin & 63;
      const float4 v4 =
          *(const float4*)(x_rec + (size_t)(rowBase + r) * TLEN + kk + j);
      *(float4*)(sA + r * 64 + j) = v4;
    }
    __syncthreads();
    #pragma unroll
    for (int ki = 0; ki < 16; ++ki) {
      int kl = ki * 4 + k0;              // K within chunk
      int kg = kk + kl;                  // global K
      v2f a, b;
      a.x = sA[lm * 64 + kl];            a.y = sA[lm * 64 + kl + 1];
      b.x = map_W[(size_t)kg * TLEN + col];
      b.y = map_W[(size_t)(kg + 1) * TLEN + col];
      acc = wmma_f32(a, b, acc);
    }
    __syncthreads();
  }

  // epilogue: + map_b, write y_hat, accumulate |y_hat - y_rec|
  float bias = map_b[col];
  float lsum = 0.0f;
  #pragma unroll
  for (int v = 0; v < 8; ++v) {
    size_t idx = (size_t)(rowBase + v + rh) * TLEN + col;
    float yh = acc[v] + bias;
    y_hat[idx] = yh;
    lsum += fabsf(yh - y_rec[idx]);
  }

  sred[tid] = lsum;
  __syncthreads();
  for (int s = 128; s > 0; s >>= 1) {
    if (tid < s) sred[tid] += sred[tid + s];
    __syncthreads();
  }
  if (tid == 0) atomicAdd(pred_acc, (double)sred[0]);
}

// ---------------------------------------------------------------------------
// Kernel 3: sum |a - b| over n elements (rec-loss numerators).
// ---------------------------------------------------------------------------
__global__ __launch_bounds__(256) void l1_reduce_kernel(
    const float* __restrict__ a, const float* __restrict__ b, int n,
    double* __restrict__ acc)
{
  __shared__ float sred[256];
  float lsum = 0.0f;
  for (int i = blockIdx.x * blockDim.x + threadIdx.x; i < n;
       i += gridDim.x * blockDim.x)
    lsum += fabsf(a[i] - b[i]);
  sred[threadIdx.x] = lsum;
  __syncthreads();
  for (int s = 128; s > 0; s >>= 1) {
    if (threadIdx.x < s) sred[threadIdx.x] += sred[threadIdx.x + s];
    __syncthreads();
  }
  if (threadIdx.x == 0) atomicAdd(acc, (double)sred[0]);
}

__global__ void zero_acc_kernel(double* acc) {
  if (threadIdx.x < 4) acc[threadIdx.x] = 0.0;
}

__global__ void finalize_kernel(const double* acc, float* out) {
  const double inv = 1.0 / (double)NTOT;
  out[0] = (float)(acc[0] * inv);                 // pred_loss
  out[1] = (float)(acc[1] * inv + acc[2] * inv);  // rec_loss
  out[2] = 0.0f;                                  // norm_loss
  out[3] = 0.0f;                                  // nonauto_loss
}

// ---------------------------------------------------------------------------
// Host launcher.  d_ws layout: [0..31]  4 doubles (loss accumulators)
//                              [256..]  x_rec (NTOT f32), then y_rec (NTOT f32)
// Required ws_size >= 256 + 2*NTOT*4  (~44 MB).
// ---------------------------------------------------------------------------
extern "C" void kernel_launch(void* const* d_in, const int* in_sizes, int n_in,
                              void* d_out, int out_size, void* d_ws, size_t ws_size,
                              hipStream_t stream) {
  const float* x      = (const float*)d_in[0];
  const float* y      = (const float*)d_in[1];
  const float* emb_W  = (const float*)d_in[2];
  const float* emb_b  = (const float*)d_in[3];
  const float* enc_W1 = (const float*)d_in[4];
  const float* enc_b1 = (const float*)d_in[5];
  const float* enc_W2 = (const float*)d_in[6];
  const float* enc_b2 = (const float*)d_in[7];
  const float* dec_W  = (const float*)d_in[8];
  const float* dec_b  = (const float*)d_in[9];
  const float* map_W  = (const float*)d_in[10];
  const float* map_b  = (const float*)d_in[11];

  float*  out   = (float*)d_out;
  float*  y_hat = out + 4;

  double* acc   = (double*)d_ws;
  float*  x_rec = (float*)((char*)d_ws + 256);
  float*  y_rec = x_rec + NTOT;

  zero_acc_kernel<<<1, 32, 0, stream>>>(acc);

  dim3 gF(BC, NMOD);
  fused_encdec_kernel<<<gF, 256, 0, stream>>>(
      x, emb_W, emb_b, enc_W1, enc_b1, enc_W2, enc_b2, dec_W, dec_b, x_rec);
  fused_encdec_kernel<<<gF, 256, 0, stream>>>(
      y, emb_W, emb_b, enc_W1, enc_b1, enc_W2, enc_b2, dec_W, dec_b, y_rec);

  dim3 gM(BC / 16, TLEN / 128);
  map_loss_kernel<<<gM, 256, 0, stream>>>(x_rec, map_W, map_b, y_rec, y_hat,
                                          acc + 0);

  l1_reduce_kernel<<<512, 256, 0, stream>>>(x_rec, x, NTOT, acc + 1);
  l1_reduce_kernel<<<512, 256, 0, stream>>>(y_rec, y, NTOT, acc + 2);

  finalize_kernel<<<1, 1, 0, stream>>>(acc, out);
}